// SampledSoftmax_51213190037826
// MI455X (gfx1250) — compile-verified
//
#include <hip/hip_runtime.h>

typedef float v2f __attribute__((ext_vector_type(2)));
typedef float v8f __attribute__((ext_vector_type(8)));

#define NTOKENS       500000
#define NSAMPLED      8192
#define NHID          256
#define BATCH         8192
#define LOGIT_STRIDE  (NSAMPLED + 1)   // 8193

// ---------------------------------------------------------------------------
// Kernel 1: true logits (gathered rowwise dot) + new_targets zeros.
// One wave32 per batch row; coalesced float4 loads + shfl_xor reduction.
// ---------------------------------------------------------------------------
__global__ void sss_true_logits(const float* __restrict__ inputs,
                                const int*   __restrict__ labels,
                                const float* __restrict__ true_log_freq,
                                const float* __restrict__ weight,
                                const float* __restrict__ bias,
                                float*       __restrict__ out) {
  const int wave = (blockIdx.x * blockDim.x + threadIdx.x) >> 5;  // 0..BATCH-1
  const int lane = threadIdx.x & 31;
  if (wave >= BATCH) return;

  const int lbl = labels[wave];
  const float4* in4 = (const float4*)(inputs + (size_t)wave * NHID);
  const float4* w4  = (const float4*)(weight + (size_t)lbl  * NHID);

  float acc = 0.0f;
#pragma unroll
  for (int i = 0; i < NHID / (32 * 4); ++i) {           // 2 iterations
    float4 a = in4[lane + 32 * i];
    float4 b = w4[lane + 32 * i];
    acc += a.x * b.x + a.y * b.y + a.z * b.z + a.w * b.w;
  }
#pragma unroll
  for (int off = 16; off > 0; off >>= 1)
    acc += __shfl_xor(acc, off, 32);

  if (lane == 0) {
    out[(size_t)wave * LOGIT_STRIDE] = acc + bias[lbl] - true_log_freq[wave];
    // new_targets[wave] = 0 (int32 zero == float 0.0f bit pattern)
    out[(size_t)BATCH * LOGIT_STRIDE + wave] = 0.0f;
  }
}

// ---------------------------------------------------------------------------
// Kernel 2: sample_logits GEMM with inline gather, f32 WMMA 16x16x4.
// Each wave32 owns a 64x64 output tile = 4x4 WMMA sub-tiles -> 128 acc VGPRs
// (fits under the launch_bounds-raised VGPR budget; verified no spills via
// disasm). Steady state per k-step: 8 global_load_b64 -> 16 v_wmma.
//
// Fragment addressing (lane = hi*16 + lo):
//   A frag (16x4): {inputs[(m0+lo)*256 + k + 2*hi], +1}
//   B frag (4x16): {weight[sid[n0+lo]*256 + k + 2*hi], +1}
//   D tile (16x16): lane, vgpr v -> D[m0 + v + 8*hi][n0 + lo]
// ---------------------------------------------------------------------------
__global__ void __launch_bounds__(256, 1)
sss_sample_gemm(const float* __restrict__ inputs,
                const int*   __restrict__ sample_ids,
                const float* __restrict__ sample_log_freq,
                const float* __restrict__ weight,
                const float* __restrict__ bias,
                float*       __restrict__ out) {
  const int lane = threadIdx.x & 31;
  const int wid  = (blockIdx.x * blockDim.x + threadIdx.x) >> 5;  // global wave id
  const int tileN = wid & 127;         // 128 tiles of 64 columns
  const int tileM = wid >> 7;          // 128 tiles of 64 rows
  const int m0 = tileM << 6;
  const int n0 = tileN << 6;
  const int hi = lane >> 4;            // 0 or 1
  const int lo = lane & 15;

  // A-row base pointers (4 M sub-tiles)
  const float* ap0 = inputs + (size_t)(m0 +  0 + lo) * NHID + 2 * hi;
  const float* ap1 = inputs + (size_t)(m0 + 16 + lo) * NHID + 2 * hi;
  const float* ap2 = inputs + (size_t)(m0 + 32 + lo) * NHID + 2 * hi;
  const float* ap3 = inputs + (size_t)(m0 + 48 + lo) * NHID + 2 * hi;

  // Gathered B-row base pointers + column corrections (4 N sub-tiles)
  const int s0 = n0 +  0 + lo, sid0 = sample_ids[s0];
  const int s1 = n0 + 16 + lo, sid1 = sample_ids[s1];
  const int s2 = n0 + 32 + lo, sid2 = sample_ids[s2];
  const int s3 = n0 + 48 + lo, sid3 = sample_ids[s3];
  const float* bp0 = weight + (size_t)sid0 * NHID + 2 * hi;
  const float* bp1 = weight + (size_t)sid1 * NHID + 2 * hi;
  const float* bp2 = weight + (size_t)sid2 * NHID + 2 * hi;
  const float* bp3 = weight + (size_t)sid3 * NHID + 2 * hi;
  const float adj0 = bias[sid0] - sample_log_freq[s0];
  const float adj1 = bias[sid1] - sample_log_freq[s1];
  const float adj2 = bias[sid2] - sample_log_freq[s2];
  const float adj3 = bias[sid3] - sample_log_freq[s3];

  const v8f zero = {0.f, 0.f, 0.f, 0.f, 0.f, 0.f, 0.f, 0.f};
  v8f c00 = zero, c01 = zero, c02 = zero, c03 = zero;
  v8f c10 = zero, c11 = zero, c12 = zero, c13 = zero;
  v8f c20 = zero, c21 = zero, c22 = zero, c23 = zero;
  v8f c30 = zero, c31 = zero, c32 = zero, c33 = zero;

#pragma unroll 2
  for (int k = 0; k < NHID; k += 4) {
    const v2f a0 = *(const v2f*)(ap0 + k);
    const v2f a1 = *(const v2f*)(ap1 + k);
    const v2f a2 = *(const v2f*)(ap2 + k);
    const v2f a3 = *(const v2f*)(ap3 + k);
    const v2f b0 = *(const v2f*)(bp0 + k);
    const v2f b1 = *(const v2f*)(bp1 + k);
    const v2f b2 = *(const v2f*)(bp2 + k);
    const v2f b3 = *(const v2f*)(bp3 + k);
    c00 = __builtin_amdgcn_wmma_f32_16x16x4_f32(false, a0, false, b0, (short)0, c00, false, false);
    c01 = __builtin_amdgcn_wmma_f32_16x16x4_f32(false, a0, false, b1, (short)0, c01, false, false);
    c02 = __builtin_amdgcn_wmma_f32_16x16x4_f32(false, a0, false, b2, (short)0, c02, false, false);
    c03 = __builtin_amdgcn_wmma_f32_16x16x4_f32(false, a0, false, b3, (short)0, c03, false, false);
    c10 = __builtin_amdgcn_wmma_f32_16x16x4_f32(false, a1, false, b0, (short)0, c10, false, false);
    c11 = __builtin_amdgcn_wmma_f32_16x16x4_f32(false, a1, false, b1, (short)0, c11, false, false);
    c12 = __builtin_amdgcn_wmma_f32_16x16x4_f32(false, a1, false, b2, (short)0, c12, false, false);
    c13 = __builtin_amdgcn_wmma_f32_16x16x4_f32(false, a1, false, b3, (short)0, c13, false, false);
    c20 = __builtin_amdgcn_wmma_f32_16x16x4_f32(false, a2, false, b0, (short)0, c20, false, false);
    c21 = __builtin_amdgcn_wmma_f32_16x16x4_f32(false, a2, false, b1, (short)0, c21, false, false);
    c22 = __builtin_amdgcn_wmma_f32_16x16x4_f32(false, a2, false, b2, (short)0, c22, false, false);
    c23 = __builtin_amdgcn_wmma_f32_16x16x4_f32(false, a2, false, b3, (short)0, c23, false, false);
    c30 = __builtin_amdgcn_wmma_f32_16x16x4_f32(false, a3, false, b0, (short)0, c30, false, false);
    c31 = __builtin_amdgcn_wmma_f32_16x16x4_f32(false, a3, false, b1, (short)0, c31, false, false);
    c32 = __builtin_amdgcn_wmma_f32_16x16x4_f32(false, a3, false, b2, (short)0, c32, false, false);
    c33 = __builtin_amdgcn_wmma_f32_16x16x4_f32(false, a3, false, b3, (short)0, c33, false, false);
  }

  // Epilogue: out[b, 1 + s], row stride 8193.
  // D tile (i,j): rows m0 + i*16 + v + 8*hi, col n0 + j*16 + lo.
  {
    const size_t row0 = (size_t)(m0 + 8 * hi);
    const size_t col  = (size_t)n0 + 1 + lo;
    float* o0 = out + (row0 +  0) * LOGIT_STRIDE + col;
    float* o1 = out + (row0 + 16) * LOGIT_STRIDE + col;
    float* o2 = out + (row0 + 32) * LOGIT_STRIDE + col;
    float* o3 = out + (row0 + 48) * LOGIT_STRIDE + col;
#pragma unroll
    for (int v = 0; v < 8; ++v) {
      const size_t r = (size_t)v * LOGIT_STRIDE;
      o0[r +  0] = c00[v] + adj0;
      o0[r + 16] = c01[v] + adj1;
      o0[r + 32] = c02[v] + adj2;
      o0[r + 48] = c03[v] + adj3;
      o1[r +  0] = c10[v] + adj0;
      o1[r + 16] = c11[v] + adj1;
      o1[r + 32] = c12[v] + adj2;
      o1[r + 48] = c13[v] + adj3;
      o2[r +  0] = c20[v] + adj0;
      o2[r + 16] = c21[v] + adj1;
      o2[r + 32] = c22[v] + adj2;
      o2[r + 48] = c23[v] + adj3;
      o3[r +  0] = c30[v] + adj0;
      o3[r + 16] = c31[v] + adj1;
      o3[r + 32] = c32[v] + adj2;
      o3[r + 48] = c33[v] + adj3;
    }
  }
}

// ---------------------------------------------------------------------------
extern "C" void kernel_launch(void* const* d_in, const int* in_sizes, int n_in,
                              void* d_out, int out_size, void* d_ws, size_t ws_size,
                              hipStream_t stream) {
  const float* inputs          = (const float*)d_in[0];
  const int*   labels          = (const int*)  d_in[1];
  const int*   sample_ids      = (const int*)  d_in[2];
  const float* true_log_freq   = (const float*)d_in[3];
  const float* sample_log_freq = (const float*)d_in[4];
  const float* weight          = (const float*)d_in[5];
  const float* bias            = (const float*)d_in[6];
  float* out = (float*)d_out;

  // Kernel 1: 8192 waves, 8 waves (256 threads) per block.
  sss_true_logits<<<BATCH / 8, 256, 0, stream>>>(
      inputs, labels, true_log_freq, weight, bias, out);

  // Kernel 2: 128 x 128 = 16384 wave-tiles, 8 waves per block -> 2048 blocks.
  sss_sample_gemm<<<(128 * 128) / 8, 256, 0, stream>>>(
      inputs, sample_ids, sample_log_freq, weight, bias, out);
}